// Segment_12120397709710
// MI455X (gfx1250) — compile-verified
//
#include <hip/hip_runtime.h>

typedef __attribute__((ext_vector_type(2))) float v2f;
typedef __attribute__((ext_vector_type(4))) float v4f;
typedef __attribute__((ext_vector_type(8))) float v8f;

// Unaligned (align-4) views for packed stores: output rows are 28B apart, so
// only 4-byte alignment can be guaranteed; gfx1250 handles unaligned b128/b64.
typedef v4f __attribute__((aligned(4))) v4f_u;
typedef v2f __attribute__((aligned(4))) v2f_u;

#define REST_ENERGY 510998.94999999994

// ---------------------------------------------------------------------------
// Stage 1: collapse the scan of drift maps. Drift couplings are additive under
// composition: composed = I with [0][1]=S, [2][3]=S, [4][5]=C where
// S = sum(lengths[:,b]) in scan order, C = -S/beta2*igamma2.
// ---------------------------------------------------------------------------
__global__ void drift_compose_kernel(const float* __restrict__ lengths, // [N,B]
                                     const float* __restrict__ energy,  // [B]
                                     float* __restrict__ sc,            // [B,2]
                                     int N, int B) {
  int b = blockIdx.x * blockDim.x + threadIdx.x;
  if (b >= B) return;
  float S = 0.0f;
  for (int n = 0; n < N; ++n) S += lengths[(size_t)n * B + b]; // scan order
  float gamma  = energy[b] / (float)REST_ENERGY;
  float ig2    = (gamma != 0.0f) ? 1.0f / (gamma * gamma) : 0.0f;
  float beta2  = 1.0f - ig2;
  float C      = -S / beta2 * ig2;
  sc[2 * b + 0] = S;
  sc[2 * b + 1] = C;
}

// Element (m,k) of the composed 7x7 map, zero-padded to 16x8.
// Column k=7 is identically zero -> B row 3 of the second WMMA may hold junk.
__device__ __forceinline__ float drift_A(int m, int k, float S, float C) {
  if (m >= 7 || k >= 7) return 0.0f;
  if (m == k) return 1.0f;
  if (m == 0 && k == 1) return S;
  if (m == 2 && k == 3) return S;
  if (m == 4 && k == 5) return C;
  return 0.0f;
}

// ---------------------------------------------------------------------------
// Stage 2: out[b,p,:] = composed @ particles[b,p,:] via V_WMMA_F32_16X16X4_F32.
// A = composed (built in registers from S,C; zero traffic, loop-invariant).
// B = 16 particles per tile (N=particle, K=coord), K=7 -> two chained WMMAs.
// D: lane n<16 holds particle n's 7 output coords in acc[0..6] -> 28
// contiguous bytes per lane, coalesced; stored non-temporally (write-once).
// Memory-bound: 224 MB @ 23.3 TB/s ~= 9.6 us floor; WMMA cost fully hidden.
// ---------------------------------------------------------------------------
__global__ void drift_apply_wmma_kernel(const float* __restrict__ particles, // [B,P,7]
                                        const float* __restrict__ sc,        // [B,2]
                                        float* __restrict__ out,             // [B,P,7]
                                        int P, int tiles) {
  const int b = blockIdx.y;
  const float S = sc[2 * b + 0];
  const float C = sc[2 * b + 1];

  const int lane = threadIdx.x & 31;
  const int half = lane >> 4;   // 0: K=0/1 (and 4/5), 1: K=2/3 (and 6/junk)
  const int m    = lane & 15;   // A row (coord) / B column (particle in tile)
  const int k0   = half * 2;
  const int k5   = k0 + 4 + (half ^ 1); // half0 -> 5 (real), half1 -> 6 (junk, killed by A col 7 == 0)

  // A operands (constant across the whole loop, zero memory traffic)
  v2f a1, a2;
  a1.x = drift_A(m, k0 + 0, S, C);
  a1.y = drift_A(m, k0 + 1, S, C);
  a2.x = drift_A(m, k0 + 4, S, C);
  a2.y = drift_A(m, k0 + 5, S, C);

  const int wavesPerBlock = blockDim.x >> 5;
  const int wave   = blockIdx.x * wavesPerBlock + ((int)threadIdx.x >> 5);
  const int nwaves = gridDim.x * wavesPerBlock;

  // Pointer-increment grid-stride loop: no per-iteration 64-bit multiply.
  const size_t stride = (size_t)nwaves * 16 * 7;          // floats per step
  const float* pp = particles + (size_t)b * P * 7 + (size_t)(wave * 16 + m) * 7;
  float*       op = out       + (size_t)b * P * 7 + (size_t)(wave * 16 + m) * 7;

  for (int t = wave; t < tiles; t += nwaves) {   // wave-uniform: EXEC all-1s
    v2f b1, b2;
    b1.x = pp[k0 + 0];
    b1.y = pp[k0 + 1];
    b2.x = pp[k0 + 4];
    b2.y = pp[k5];                 // branch-free, always in-bounds

    v8f acc = {};
    acc = __builtin_amdgcn_wmma_f32_16x16x4_f32(false, a1, false, b1,
                                                (short)0, acc, false, false);
    acc = __builtin_amdgcn_wmma_f32_16x16x4_f32(false, a2, false, b2,
                                                (short)0, acc, false, false);

    if (half == 0) {               // lanes 0..15 own particles' outputs
      v4f w4; w4.x = acc[0]; w4.y = acc[1]; w4.z = acc[2]; w4.w = acc[3];
      v2f w2; w2.x = acc[4]; w2.y = acc[5];
      __builtin_nontemporal_store(w4, (v4f_u*)(op + 0));
      __builtin_nontemporal_store(w2, (v2f_u*)(op + 4));
      __builtin_nontemporal_store(acc[6], op + 6);
    }
    pp += stride;
    op += stride;
  }
}

// Scalar tail for P % 16 != 0 (P = 1e6 is exact, but stay general).
__global__ void drift_apply_tail_kernel(const float* __restrict__ particles,
                                        const float* __restrict__ sc,
                                        float* __restrict__ out,
                                        int P, int start) {
  const int b = blockIdx.y;
  const int p = start + blockIdx.x * blockDim.x + threadIdx.x;
  if (p >= P) return;
  const float S = sc[2 * b + 0];
  const float C = sc[2 * b + 1];
  const float* pp = particles + ((size_t)b * P + p) * 7;
  float*       op = out       + ((size_t)b * P + p) * 7;
  float x0 = pp[0], x1 = pp[1], x2 = pp[2], x3 = pp[3];
  float x4 = pp[4], x5 = pp[5], x6 = pp[6];
  op[0] = fmaf(S, x1, x0);
  op[1] = x1;
  op[2] = fmaf(S, x3, x2);
  op[3] = x3;
  op[4] = fmaf(C, x5, x4);
  op[5] = x5;
  op[6] = x6;
}

extern "C" void kernel_launch(void* const* d_in, const int* in_sizes, int n_in,
                              void* d_out, int out_size, void* d_ws, size_t ws_size,
                              hipStream_t stream) {
  const float* lengths   = (const float*)d_in[0]; // [N,B]
  const float* energy    = (const float*)d_in[1]; // [B]
  const float* particles = (const float*)d_in[2]; // [B,P,7]
  float* out = (float*)d_out;
  float* sc  = (float*)d_ws;                      // [B,2] scratch

  const int B = in_sizes[1];
  const int N = in_sizes[0] / B;
  const int P = (in_sizes[2] / B) / 7;
  const int tiles = P / 16;
  const int tail  = P - tiles * 16;

  drift_compose_kernel<<<1, 32, 0, stream>>>(lengths, energy, sc, N, B);

  dim3 grid(512, B);
  drift_apply_wmma_kernel<<<grid, 256, 0, stream>>>(particles, sc, out, P, tiles);

  if (tail > 0) {
    dim3 tgrid((tail + 255) / 256, B);
    drift_apply_tail_kernel<<<tgrid, 256, 0, stream>>>(particles, sc, out,
                                                       P, tiles * 16);
  }
}